// D_SUM_CALC_29987461660959
// MI455X (gfx1250) — compile-verified
//
#include <hip/hip_runtime.h>

#define NN 4096
#define TILES (NN / 16)

typedef __attribute__((ext_vector_type(2))) float v2f;
typedef __attribute__((ext_vector_type(8))) float v8f;

// ---------------------------------------------------------------------------
// Phase 1: row-wise inclusive prefix sum, D -> R (in d_ws).
// One wave32 per 16-row band. Per 16x16 tile: P = T x U (U = upper-tri ones),
// done as 4x V_WMMA_F32_16X16X4_F32, then add per-row carry, store, update
// carry by broadcasting column 15 (lane 15 / lane 31) to the half-wave.
// ---------------------------------------------------------------------------
__global__ __launch_bounds__(256) void row_scan_wmma(const float* __restrict__ D,
                                                     float* __restrict__ R) {
    const int lane = threadIdx.x & 31;
    const int wave = threadIdx.x >> 5;
    const int band = blockIdx.x * 8 + wave;   // 256 bands of 16 rows
    const int row0 = band * 16;
    const int rl   = lane & 15;               // A: row M ; C/D: col N
    const int half = lane >> 4;

    // B operand: U[k][n] = (k <= n), chunked into 4 K=4 slices.
    // B layout: lane l holds N = l&15, K(in-chunk) = 2*half + v.
    v2f u[4];
#pragma unroll
    for (int j = 0; j < 4; ++j) {
        const int k0 = 4 * j + 2 * half;
        u[j].x = (k0     <= rl) ? 1.0f : 0.0f;
        u[j].y = (k0 + 1 <= rl) ? 1.0f : 0.0f;
    }

    v8f carry = {};
    const float* arow = D + (size_t)(row0 + rl) * NN;  // A: lane's row
    float*       orow = R;

    for (int t = 0; t < TILES; ++t) {
        const int c0 = t * 16;
        // WGP-scope prefetch (high locality), two tiles ahead to cover HBM
        // latency behind the serial WMMA chain.
        if (t + 2 < TILES)
            __builtin_prefetch(arow + c0 + 32, 0, 3);  // global_prefetch_b8, WGP scope

        // A chunks: lane l holds T[l&15, c0 + 4j + 2*half + {0,1}] (8B aligned)
        v8f acc = {};
#pragma unroll
        for (int j = 0; j < 4; ++j) {
            v2f a = *(const v2f*)(arow + c0 + 4 * j + 2 * half);
            acc = __builtin_amdgcn_wmma_f32_16x16x4_f32(
                false, a, false, u[j], (short)0, acc, false, false);
        }

        acc = acc + carry;   // carry[p] is uniform per half-wave = row running sum

#pragma unroll
        for (int p = 0; p < 8; ++p)
            orow[(size_t)(row0 + p + 8 * half) * NN + c0 + rl] = acc[p];

        // new carry = column 15 of this tile (lane 15 rows 0..7, lane 31 rows 8..15)
        const int src = 15 + (half << 4);
#pragma unroll
        for (int p = 0; p < 8; ++p)
            carry[p] = __shfl(acc[p], src, 32);
    }
}

// ---------------------------------------------------------------------------
// Phase 2: column-wise inclusive prefix sum, in place on R (-> S).
// One wave32 per 16-column band. Per tile: P = L x R (L = lower-tri ones),
// 4x WMMA; carry is one value per column (single VGPR, replicated halves),
// updated by broadcasting row 15 (acc[7], lanes 16..31).
// ---------------------------------------------------------------------------
__global__ __launch_bounds__(256) void col_scan_wmma(float* __restrict__ S) {
    const int lane = threadIdx.x & 31;
    const int wave = threadIdx.x >> 5;
    const int band = blockIdx.x * 8 + wave;   // 256 bands of 16 columns
    const int col0 = band * 16;
    const int cl   = lane & 15;               // A: row M ; B/C/D: col N
    const int half = lane >> 4;

    // A operand: L[m][k] = (k <= m), lane holds M = l&15, K = 4j + 2*half + v.
    v2f a[4];
#pragma unroll
    for (int j = 0; j < 4; ++j) {
        const int k0 = 4 * j + 2 * half;
        a[j].x = (k0     <= cl) ? 1.0f : 0.0f;
        a[j].y = (k0 + 1 <= cl) ? 1.0f : 0.0f;
    }

    float creg = 0.0f;  // per-column carry, lane l holds carry[col0 + (l&15)]

    for (int t = 0; t < TILES; ++t) {
        const int r0 = t * 16;
        // WGP-scope prefetch of the tile after next (one row per half-wave,
        // rows r0+32 and r0+34, 2 of the future tile's 16 rows).
        if (t + 2 < TILES)
            __builtin_prefetch(S + (size_t)(r0 + 32 + 2 * half) * NN + col0 + cl, 0, 3);

        v8f acc = {};
#pragma unroll
        for (int j = 0; j < 4; ++j) {
            const int kr = r0 + 4 * j + 2 * half;   // B: K rows, coalesced per row
            v2f b;
            b.x = S[(size_t)kr * NN + col0 + cl];
            b.y = S[(size_t)(kr + 1) * NN + col0 + cl];
            acc = __builtin_amdgcn_wmma_f32_16x16x4_f32(
                false, a[j], false, b, (short)0, acc, false, false);
        }

#pragma unroll
        for (int p = 0; p < 8; ++p)
            acc[p] += creg;

#pragma unroll
        for (int p = 0; p < 8; ++p)
            S[(size_t)(r0 + p + 8 * half) * NN + col0 + cl] = acc[p];

        // new carry = row 15 of this tile = acc[7] in lanes 16..31 (M = 15)
        creg = __shfl(acc[7], 16 + cl, 32);
    }
}

// ---------------------------------------------------------------------------
// Phase 3: inclusion-exclusion gather (S stays L2-resident: 64MB < 192MB L2).
// out[r,c] = S[J,J] - S[I-1,J] - S[J,I-1] + S[I-1,I-1]; subdiagonal quirk.
// ---------------------------------------------------------------------------
__global__ __launch_bounds__(256) void gather_out(const float* __restrict__ S,
                                                  const float* __restrict__ D,
                                                  float* __restrict__ out) {
    const int c = blockIdx.x * 32 + (threadIdx.x & 31);
    const int r = blockIdx.y * 8 + (threadIdx.x >> 5);
    const int I = (r < c) ? r : c;
    const int J = (r < c) ? c : r;

    float s = S[(size_t)J * NN + J];
    if (I > 0) {
        s -= S[(size_t)(I - 1) * NN + J];
        s -= S[(size_t)J * NN + (I - 1)];
        s += S[(size_t)(I - 1) * NN + (I - 1)];
    }
    if (r == c + 1)                       // quirk: out[i+1,i] = D[i,i+1]
        s = D[(size_t)c * NN + r];

    out[(size_t)r * NN + c] = s;
}

extern "C" void kernel_launch(void* const* d_in, const int* in_sizes, int n_in,
                              void* d_out, int out_size, void* d_ws, size_t ws_size,
                              hipStream_t stream) {
    (void)in_sizes; (void)n_in; (void)out_size; (void)ws_size;

    const float* D = (const float*)d_in[0];     // [1, N, N] f32
    float*       O = (float*)d_out;             // [1, N, N] f32
    float*       S = (float*)d_ws;              // N*N f32 scratch (64 MB)

    row_scan_wmma<<<TILES / 8, 256, 0, stream>>>(D, S);       // 32 blocks x 8 waves
    col_scan_wmma<<<TILES / 8, 256, 0, stream>>>(S);          // in-place column scan
    gather_out<<<dim3(NN / 32, NN / 8), 256, 0, stream>>>(S, D, O);
}